// MDE_Module_27951647162478
// MI455X (gfx1250) — compile-verified
//
#include <hip/hip_runtime.h>
#include <hip/hip_bf16.h>

typedef __attribute__((ext_vector_type(2))) float v2f;
typedef __attribute__((ext_vector_type(8))) float v8f;

// x: [8,32,32,32,32] f32 row-major. 262144 spatial rows x 32 features.
// out: same rows x 32 outputs.
// Strides in floats: f:1, k:32, j:1024, i:32768, b:1048576.

// ---------------- first-level mean reductions (read x once each) -------------

__global__ void reduce_i_kernel(const float* __restrict__ x, float* __restrict__ m1) {
  int t = blockIdx.x * blockDim.x + threadIdx.x;   // (b,j,k,f) -> 262144 threads
  int jkf = t & 32767;
  int b   = t >> 15;
  const float* p = x + (size_t)b * 1048576 + jkf;
  float s = 0.f;
#pragma unroll
  for (int i = 0; i < 32; ++i) s += p[(size_t)i * 32768];
  m1[t] = s * (1.f / 32.f);
}

__global__ void reduce_j_kernel(const float* __restrict__ x, float* __restrict__ m2) {
  int t = blockIdx.x * blockDim.x + threadIdx.x;   // (b,i,k,f)
  int kf = t & 1023;
  int bi = t >> 10;
  const float* p = x + (size_t)bi * 32768 + kf;
  float s = 0.f;
#pragma unroll
  for (int j = 0; j < 32; ++j) s += p[(size_t)j * 1024];
  m2[t] = s * (1.f / 32.f);
}

__global__ void reduce_k_kernel(const float* __restrict__ x, float* __restrict__ m3) {
  int t = blockIdx.x * blockDim.x + threadIdx.x;   // (b,i,j,f)
  int f   = t & 31;
  int bij = t >> 5;
  const float* p = x + (size_t)bij * 1024 + f;
  float s = 0.f;
#pragma unroll
  for (int k = 0; k < 32; ++k) s += p[k * 32];
  m3[t] = s * (1.f / 32.f);
}

// ------------- second-level means over the small tensors ---------------------
// dst[t] with t = (b, mid, f); src index = b*b_stride + mid*mid_stride + f,
// reduce 32 values spaced red_stride apart, then /32.
__global__ void mean32_kernel(const float* __restrict__ src, float* __restrict__ dst,
                              int n, int b_stride, int mid_stride, int red_stride) {
  int t = blockIdx.x * blockDim.x + threadIdx.x;
  if (t >= n) return;
  int f   = t & 31;
  int mid = (t >> 5) & 31;
  int b   = t >> 10;
  const float* p = src + (size_t)b * b_stride + (size_t)mid * mid_stride + f;
  float s = 0.f;
#pragma unroll
  for (int r = 0; r < 32; ++r) s += p[(size_t)r * red_stride];
  dst[t] = s * (1.f / 32.f);
}

// ------------- build additive gather tensors Aadd/Badd/Cadd ------------------
// t = (pos, o); pos = (b, p, q). Interpreting pos three ways simultaneously:
//   Aadd: pos=(b,j,k)  uses m1[pos], m12[b,q], m13[b,p], m123[b], bias
//   Badd: pos=(b,i,k)  uses m2[pos], m23[b,p]
//   Cadd: pos=(b,i,j)  uses m3[pos]
__global__ void build_add_kernel(const float* __restrict__ W, const float* __restrict__ bias,
                                 const float* __restrict__ m1, const float* __restrict__ m12,
                                 const float* __restrict__ m13, const float* __restrict__ m123,
                                 const float* __restrict__ m2, const float* __restrict__ m23,
                                 const float* __restrict__ m3,
                                 float* __restrict__ Aadd, float* __restrict__ Badd,
                                 float* __restrict__ Cadd) {
  int t = blockIdx.x * blockDim.x + threadIdx.x;   // 262144*? no: 262144 pos *32 o /32 -> t over 262144*32? No: t over pos*32
  int o   = t & 31;
  int pos = t >> 5;
  int q = pos & 31;
  int p = (pos >> 5) & 31;
  int b = pos >> 10;
  const float* Wr = W + (size_t)o * 256;
  const float* v1   = m1   + (size_t)pos * 32;
  const float* v12  = m12  + ((size_t)b * 32 + q) * 32;
  const float* v13  = m13  + ((size_t)b * 32 + p) * 32;
  const float* v123 = m123 + (size_t)b * 32;
  const float* v2   = m2   + (size_t)pos * 32;
  const float* v23  = m23  + ((size_t)b * 32 + p) * 32;
  const float* v3   = m3   + (size_t)pos * 32;
  float a = bias[o], bs = 0.f, cs = 0.f;
#pragma unroll
  for (int f = 0; f < 32; ++f) {
    a  += v1[f] * Wr[32 + f] + v12[f] * Wr[128 + f] + v13[f] * Wr[160 + f] + v123[f] * Wr[224 + f];
    bs += v2[f] * Wr[64 + f] + v23[f] * Wr[192 + f];
    cs += v3[f] * Wr[96 + f];
  }
  Aadd[t] = a; Badd[t] = bs; Cadd[t] = cs;
}

// ------------- main fused GEMM: out = x·W0^T + gathers ----------------------
// One wave per 16-row x 32-col output tile; V_WMMA_F32_16X16X4_F32 chained 8x
// (K=32) for each of the two 16x16 N-halves. 4096 waves x 4 tiles = 262144 rows.
__global__ void __launch_bounds__(256) main_gemm_kernel(
    const float* __restrict__ x, const float* __restrict__ W,
    const float* __restrict__ Aadd, const float* __restrict__ Badd,
    const float* __restrict__ Cadd, float* __restrict__ out) {
  const int lane  = threadIdx.x & 31;
  const int wave  = threadIdx.x >> 5;
  const int waveId = blockIdx.x * 8 + wave;        // 0..4095
  const int mlane = lane & 15;                     // M (A) / N (B,D) index
  const int khalf = lane >> 4;                     // selects K-pair (A/B) / M+8 (D)

  // Preload B fragments: bfrag[h][s] holds B[k][n] = W[n, k] for n-half h,
  // k-step s. ISA B layout (4x16 f32): VGPR0 lanes0-15 row K=0 / lanes16-31
  // row K=2; VGPR1 rows K=1/K=3.
  v2f bfrag[2][8];
#pragma unroll
  for (int h = 0; h < 2; ++h) {
    const float* Wr = W + (size_t)(mlane + 16 * h) * 256;
#pragma unroll
    for (int s = 0; s < 8; ++s) {
      int kk = 4 * s + 2 * khalf;
      v2f bv; bv.x = Wr[kk]; bv.y = Wr[kk + 1];
      bfrag[h][s] = bv;
    }
  }

  for (int rep = 0; rep < 4; ++rep) {
    const int tile = waveId + rep * 4096;          // 0..16383
    const int row0 = tile * 16;
    const int k0 = row0 & 31;                      // 0 or 16 (tile within k-dim)
    const int j  = (row0 >> 5)  & 31;
    const int i  = (row0 >> 10) & 31;
    const int b  =  row0 >> 15;

    // A fragments: lane holds A[M=mlane][K = 4s + 2*khalf + {0,1}]
    const float* xrow = x + (size_t)(row0 + mlane) * 32;
    v2f afrag[8];
#pragma unroll
    for (int s = 0; s < 8; ++s)
      afrag[s] = *(const v2f*)(xrow + 4 * s + 2 * khalf);

    const float* Abase = Aadd + (((size_t)(b * 32 + j) * 32 + k0) * 32);
    const float* Bbase = Badd + (((size_t)(b * 32 + i) * 32 + k0) * 32);
    const float* Cbase = Cadd + (((size_t)(b * 32 + i) * 32 + j) * 32);

#pragma unroll
    for (int h = 0; h < 2; ++h) {
      v8f acc = {};
#pragma unroll
      for (int s = 0; s < 8; ++s) {
        acc = __builtin_amdgcn_wmma_f32_16x16x4_f32(
            /*neg_a=*/false, afrag[s], /*neg_b=*/false, bfrag[h][s],
            /*c_mod=*/(short)0, acc, /*reuse_a=*/false, /*reuse_b=*/false);
      }
      const int ncol = mlane + 16 * h;             // output channel
#pragma unroll
      for (int v = 0; v < 8; ++v) {
        const int M = v + 8 * khalf;               // row-in-tile = k offset
        float add = Abase[(size_t)M * 32 + ncol] +
                    Bbase[(size_t)M * 32 + ncol] +
                    Cbase[ncol];
        out[(size_t)(row0 + M) * 32 + ncol] = acc[v] + add;
      }
    }
  }
}

// ---------------------------------------------------------------------------

extern "C" void kernel_launch(void* const* d_in, const int* in_sizes, int n_in,
                              void* d_out, int out_size, void* d_ws, size_t ws_size,
                              hipStream_t stream) {
  const float* x    = (const float*)d_in[0];   // 8*32*32*32*32
  const float* W    = (const float*)d_in[1];   // 32 x 256
  const float* bias = (const float*)d_in[2];   // 32
  float* out = (float*)d_out;                  // 262144 x 32
  float* ws  = (float*)d_ws;

  // workspace layout (floats); total 1,597,696 floats (~6.1 MB)
  float* m1   = ws;                 // (b,j,k,f) 262144
  float* m2   = ws + 262144;        // (b,i,k,f) 262144
  float* m3   = ws + 524288;        // (b,i,j,f) 262144
  float* m12  = ws + 786432;        // (b,k,f)   8192
  float* m13  = ws + 794624;        // (b,j,f)   8192
  float* m23  = ws + 802816;        // (b,i,f)   8192
  float* m123 = ws + 811008;        // (b,f)     256
  float* Aadd = ws + 811264;        // (b,j,k,o) 262144
  float* Badd = ws + 1073408;       // (b,i,k,o) 262144
  float* Cadd = ws + 1335552;       // (b,i,j,o) 262144

  // first-level means
  reduce_i_kernel<<<1024, 256, 0, stream>>>(x, m1);
  reduce_j_kernel<<<1024, 256, 0, stream>>>(x, m2);
  reduce_k_kernel<<<1024, 256, 0, stream>>>(x, m3);

  // second-level means
  mean32_kernel<<<32, 256, 0, stream>>>(m1,  m12,  8192, 32768, 32,   1024); // mean_j m1
  mean32_kernel<<<32, 256, 0, stream>>>(m1,  m13,  8192, 32768, 1024, 32);   // mean_k m1
  mean32_kernel<<<32, 256, 0, stream>>>(m3,  m23,  8192, 32768, 1024, 32);   // mean_j m3
  mean32_kernel<<<1,  256, 0, stream>>>(m13, m123, 256,  0,     1024, 32);   // mean_j m13

  // additive gather tensors
  build_add_kernel<<<32768, 256, 0, stream>>>(W, bias, m1, m12, m13, m123,
                                              m2, m23, m3, Aadd, Badd, Cadd);

  // fused WMMA GEMM + gathers
  main_gemm_kernel<<<512, 256, 0, stream>>>(x, W, Aadd, Badd, Cadd, out);
}